// DeformableConv_18983755448465
// MI455X (gfx1250) — compile-verified
//
#include <hip/hip_runtime.h>
#include <hip/hip_bf16.h>

typedef __attribute__((ext_vector_type(2))) float v2f;
typedef __attribute__((ext_vector_type(8))) float v8f;

// Problem constants (match reference)
constexpr int Bv = 4, Cv = 256, Hv = 64, Wv = 64;
constexpr int OC = 256, TAPS = 9;            // 3x3 kernel
constexpr int HoW = 64 * 64;                 // Ho*Wo per image = 4096
constexpr int Mtot = Bv * HoW;               // 16384 GEMM rows

// Tiling
constexpr int BM = 64;                        // rows per block (32 per wave)
constexpr int CC = 4;                         // channels per K-chunk
constexpr int KC = CC * TAPS;                 // 36 k per chunk
constexpr int AP = 38;                        // A row stride: even (b64 pairs), 19 coprime 32
constexpr int KP = 38;                        // B row stride (Bs[oc][kk]), same property

__global__ __launch_bounds__(256)
void dcn_fused_wmma(const float* __restrict__ x,
                    const float* __restrict__ off,
                    const float* __restrict__ wgt,
                    float* __restrict__ out)
{
    __shared__ __align__(16) float Bs[OC * KP];   // Wt[oc][kk]    ~38.9 KB
    __shared__ __align__(16) float As[BM * AP];   // cols[row][kk]  ~9.7 KB
    __shared__ int   mIdx[BM * TAPS][4];          // clamped corner indices  ~9.2 KB
    __shared__ float mWt [BM * TAPS][4];          // bilinear weights (0 if OOB)

    const int t    = threadIdx.x;
    const int lane = t & 31;
    const int wave = t >> 5;
    const int m0   = blockIdx.x * BM;

    const int wm = wave >> 2;               // 0..1 : M half (32 rows each)
    const int wn = wave & 3;                // 0..3 : N sub-tile (64 oc each)
    const int lg = lane >> 4;               // lane group -> K offset 0 / 2
    const int ln = lane & 15;               // fragment row/col index

    // ---- Per-row/per-tap sampling metadata (shared across all 256 channels) ----
    for (int e = t; e < BM * TAPS; e += 256) {
        const int row = e / TAPS, tap = e % TAPS;
        const int m   = m0 + row;
        const int b   = m >> 12;
        const int pix = m & 4095;
        const int ho  = pix >> 6, wo = pix & 63;
        const int ki  = tap / 3,  kj = tap % 3;
        const float dy = off[(( (size_t)b * 18 + 2 * tap    ) << 12) + pix];
        const float dx = off[(( (size_t)b * 18 + 2 * tap + 1) << 12) + pix];
        const float ph = (float)(ho - 1 + ki) + dy;   // stride 1, pad 1, dil 1
        const float pw = (float)(wo - 1 + kj) + dx;
        const float h0f = floorf(ph), w0f = floorf(pw);
        const float lh = ph - h0f, lw = pw - w0f;
        const int h0 = (int)h0f, w0 = (int)w0f;
#pragma unroll
        for (int c4 = 0; c4 < 4; ++c4) {
            const int di = c4 >> 1, dj = c4 & 1;
            const int hh = h0 + di, ww = w0 + dj;
            const bool valid = (hh >= 0) & (hh < Hv) & (ww >= 0) & (ww < Wv);
            const float wtv = (di ? lh : 1.0f - lh) * (dj ? lw : 1.0f - lw);
            const int hc = min(max(hh, 0), Hv - 1);
            const int wc = min(max(ww, 0), Wv - 1);
            mIdx[e][c4] = hc * Wv + wc;               // clamped; weight 0 if OOB
            mWt [e][c4] = valid ? wtv : 0.0f;
        }
    }

    v8f acc[2][4] = {};                      // 32x64 fp32 per wave

    for (int c0 = 0; c0 < Cv; c0 += CC) {
        __syncthreads();                     // As/Bs safe to overwrite

        // ---- stage weight chunk transposed: Bs[oc][kk], thread t <-> oc ----
        {
            const int oc = t;
            const float4* wsrc =
                (const float4*)(wgt + (size_t)oc * (Cv * TAPS) + c0 * TAPS);
            if (c0 + CC < Cv)
                __builtin_prefetch(wgt + (size_t)oc * (Cv * TAPS) + (c0 + CC) * TAPS, 0, 3);
#pragma unroll
            for (int q = 0; q < 9; ++q) {
                const float4 v = wsrc[q];
                const int base = oc * KP + q * 4;     // even -> 8B-aligned pairs
                *(v2f*)&Bs[base + 0] = v2f{v.x, v.y};
                *(v2f*)&Bs[base + 2] = v2f{v.z, v.w};
            }
        }

        // ---- build A tile: bilinear gather, 64 rows x 36 kk ----
        for (int e = t; e < BM * KC; e += 256) {
            const int row = e / KC, kk = e % KC;
            const int c   = c0 + kk / TAPS;
            const int tap = kk % TAPS;
            const int m   = m0 + row;
            const int b   = m >> 12;
            const float* xp = x + ((size_t)(b * Cv + c) << 12);
            const int*   id = mIdx[row * TAPS + tap];
            const float* wv = mWt [row * TAPS + tap];
            As[row * AP + kk] = wv[0] * xp[id[0]] + wv[1] * xp[id[1]]
                              + wv[2] * xp[id[2]] + wv[3] * xp[id[3]];
        }
        __syncthreads();

        // ---- fp32 WMMA contraction: 9 K-steps, 8 WMMAs per step ----
#pragma unroll
        for (int ks = 0; ks < 9; ++ks) {
            const int kb = ks * 4 + lg * 2;          // even -> aligned b64 loads
            const v2f a0 = *(const v2f*)&As[(wm * 32 +      ln) * AP + kb];
            const v2f a1 = *(const v2f*)&As[(wm * 32 + 16 + ln) * AP + kb];
#pragma unroll
            for (int nt = 0; nt < 4; ++nt) {
                const int oc = wn * 64 + nt * 16 + ln;
                const v2f bf = *(const v2f*)&Bs[oc * KP + kb];
                acc[0][nt] = __builtin_amdgcn_wmma_f32_16x16x4_f32(
                    false, a0, false, bf, (short)0, acc[0][nt], false, false);
                acc[1][nt] = __builtin_amdgcn_wmma_f32_16x16x4_f32(
                    false, a1, false, bf, (short)0, acc[1][nt], false, false);
            }
        }
    }

    // ---- epilogue: D layout VGPR v -> M = v + 8*lg, N = ln ----
#pragma unroll
    for (int sub = 0; sub < 2; ++sub) {
#pragma unroll
        for (int nt = 0; nt < 4; ++nt) {
            const int oc = wn * 64 + nt * 16 + ln;
#pragma unroll
            for (int v = 0; v < 8; ++v) {
                const int m   = m0 + wm * 32 + sub * 16 + v + 8 * lg;
                const int b   = m >> 12;
                const int pix = m & 4095;
                out[(((size_t)b * OC + oc) << 12) + pix] = acc[sub][nt][v];
            }
        }
    }
}

extern "C" void kernel_launch(void* const* d_in, const int* in_sizes, int n_in,
                              void* d_out, int out_size, void* d_ws, size_t ws_size,
                              hipStream_t stream) {
    const float* x   = (const float*)d_in[0];
    const float* off = (const float*)d_in[1];
    const float* wgt = (const float*)d_in[2];
    float* out = (float*)d_out;
    (void)in_sizes; (void)n_in; (void)out_size; (void)d_ws; (void)ws_size;

    dim3 grid(Mtot / BM);   // 256 blocks
    dim3 block(256);        // 8 wave32
    dcn_fused_wmma<<<grid, block, 0, stream>>>(x, off, wgt, out);
}